// Attention_5308579577992
// MI455X (gfx1250) — compile-verified
//
#include <hip/hip_runtime.h>
#include <hip/hip_bf16.h>
#include <stdint.h>

// Problem constants (match reference)
#define BB 2
#define NN 2048
#define DD 1024
#define HH 16
#define DH 64
#define INNER 1024
#define EPS 1e-5f
#define MASK_VAL (-3.4028234663852886e+38f)

typedef __attribute__((ext_vector_type(2))) float v2f;
typedef __attribute__((ext_vector_type(8))) float v8f;

// ---------------------------------------------------------------------------
// Kernel 1: LayerNorm  xn = (x - mu)/sqrt(var+eps)*w + b    (one block per row)
// ---------------------------------------------------------------------------
__global__ void ln_kernel(const float* __restrict__ x,
                          const float* __restrict__ w,
                          const float* __restrict__ bia,
                          float* __restrict__ xn) {
    const int row = blockIdx.x;
    const float* xr = x + (size_t)row * DD;
    __shared__ float s1[256];
    __shared__ float s2[256];
    float sum = 0.f, sum2 = 0.f;
    for (int i = threadIdx.x; i < DD; i += 256) {
        float v = xr[i];
        sum += v;
        sum2 += v * v;
    }
    s1[threadIdx.x] = sum;
    s2[threadIdx.x] = sum2;
    __syncthreads();
    for (int off = 128; off > 0; off >>= 1) {
        if (threadIdx.x < off) {
            s1[threadIdx.x] += s1[threadIdx.x + off];
            s2[threadIdx.x] += s2[threadIdx.x + off];
        }
        __syncthreads();
    }
    const float mu = s1[0] * (1.0f / DD);
    const float var = s2[0] * (1.0f / DD) - mu * mu;
    const float rstd = rsqrtf(var + EPS);
    float* xnr = xn + (size_t)row * DD;
    for (int i = threadIdx.x; i < DD; i += 256) {
        xnr[i] = (xr[i] - mu) * rstd * w[i] + bia[i];
    }
}

// ---------------------------------------------------------------------------
// Kernel 2: generic fp32 WMMA GEMM   C[M x Nc] = scale * (A[M x K] @ B[K x Nc])
// Block = 256 threads = 8 waves; each wave owns one 16x16 tile; block covers
// 16 rows x 128 cols. grid = (M/16, Nc/128). wave32 WMMA f32 16x16x4.
// ---------------------------------------------------------------------------
__global__ void gemm16_kernel(const float* __restrict__ A,
                              const float* __restrict__ Bm,
                              float* __restrict__ C,
                              int M, int K, int Nc, float scale) {
    const int wave = threadIdx.x >> 5;
    const int lane = threadIdx.x & 31;
    const int half = lane >> 4;  // 0: lanes 0-15, 1: lanes 16-31
    const int l16  = lane & 15;

    const int m0 = blockIdx.x * 16;
    const int n0 = blockIdx.y * 128 + wave * 16;

    const float* ap = A + (size_t)(m0 + l16) * K + half * 2;
    const float* bp = Bm + (size_t)(half * 2) * Nc + n0 + l16;

    v8f acc = {};
#pragma unroll 4
    for (int k0 = 0; k0 < K; k0 += 4) {
        // A-matrix 16x4: lane<16 holds (K=k0,k0+1), lane>=16 holds (K=k0+2,k0+3)
        v2f a = *(const v2f*)(ap + k0);
        // B-matrix 4x16: VGPR0 row k0 (+2 hi half), VGPR1 row k0+1 (+3 hi half)
        const float* bpk = bp + (size_t)k0 * Nc;
        v2f b;
        b.x = bpk[0];
        b.y = bpk[Nc];
        acc = __builtin_amdgcn_wmma_f32_16x16x4_f32(
            false, a, false, b, (short)0, acc, false, false);
    }
#pragma unroll
    for (int r = 0; r < 8; ++r) {
        const int row = m0 + r + half * 8;
        C[(size_t)row * Nc + n0 + l16] = acc[r] * scale;
    }
}

// ---------------------------------------------------------------------------
// Kernel 3: fused flash attention, fp32, one wave per (b, h, 16-row i-tile).
//   sim = q@k^T + bias, mask -> -FLT_MAX, online softmax, O += P@V.
// q: [B*N, INNER] (pre-scaled), kv: [B*N, 128] (k = [:,:64], v = [:,64:]).
// attn_bias streamed with non-temporal loads + prefetch (268 MB, HBM-bound).
// ---------------------------------------------------------------------------
__global__ __launch_bounds__(32)
void attn_kernel(const float* __restrict__ q,
                 const float* __restrict__ kv,
                 const float* __restrict__ attn_bias,
                 const int* __restrict__ mask,
                 float* __restrict__ out_tmp) {
    const int itile = blockIdx.x;
    const int h = blockIdx.y;
    const int b = blockIdx.z;
    const int i0 = itile * 16;

    const int lane = threadIdx.x & 31;
    const int half = lane >> 4;
    const int l16  = lane & 15;

    __shared__ float pLDS[16 * 16];

    // Q tile in A-layout registers: A[m][k], m = l16, k = kk*4 + half*2 + {0,1}
    const float* qbase = q + ((size_t)(b * NN) + i0 + l16) * INNER + h * DH;
    v2f qreg[16];
#pragma unroll
    for (int kk = 0; kk < 16; ++kk)
        qreg[kk] = *(const v2f*)(qbase + kk * 4 + half * 2);

    // Per-lane online-softmax state for rows (r + 8*half), duplicated in 16-lane group
    float mrow[8], lrow[8];
    v8f o[4];
#pragma unroll
    for (int r = 0; r < 8; ++r) { mrow[r] = -INFINITY; lrow[r] = 0.f; }
#pragma unroll
    for (int t = 0; t < 4; ++t) o[t] = (v8f){};

    const float* kv_b = kv + (size_t)(b * NN) * 128;

    for (int j0 = 0; j0 < NN; j0 += 16) {
        // ---- S = Q @ K^T (16x16), K-dim = 64 ----
        v8f s = {};
        const float* krow = kv_b + (size_t)(j0 + l16) * 128 + half * 2;
#pragma unroll
        for (int kk = 0; kk < 16; ++kk) {
            v2f bvec = *(const v2f*)(krow + kk * 4);
            s = __builtin_amdgcn_wmma_f32_16x16x4_f32(
                false, qreg[kk], false, bvec, (short)0, s, false, false);
        }

        // ---- bias + mask (bias is the HBM-bound stream: NT loads + prefetch) ----
        const float* biasp = attn_bias + ((size_t)h * NN + i0) * NN + j0 + l16;
        if (j0 + 32 < NN)
            __builtin_prefetch(biasp + 32, 0, 0);
        const bool ok = mask[b * NN + j0 + l16] != 0;
#pragma unroll
        for (int r = 0; r < 8; ++r) {
            const float bias = __builtin_nontemporal_load(
                biasp + (size_t)(r + half * 8) * NN);
            s[r] = ok ? (s[r] + bias) : MASK_VAL;
        }

        // ---- online softmax: reductions across 16-lane groups (N = lane) ----
        float alpha[8];
#pragma unroll
        for (int r = 0; r < 8; ++r) {
            float v = s[r];
#pragma unroll
            for (int off = 1; off < 16; off <<= 1)
                v = fmaxf(v, __shfl_xor(v, off, 32));
            const float mnew = fmaxf(mrow[r], v);
            alpha[r] = __expf(mrow[r] - mnew);
            mrow[r] = mnew;
            const float p = __expf(s[r] - mnew);
            s[r] = p;
            float psum = p;
#pragma unroll
            for (int off = 1; off < 16; off <<= 1)
                psum += __shfl_xor(psum, off, 32);
            lrow[r] = lrow[r] * alpha[r] + psum;
        }

        // rescale accumulated O
#pragma unroll
        for (int t = 0; t < 4; ++t)
#pragma unroll
            for (int r = 0; r < 8; ++r)
                o[t][r] *= alpha[r];

        // ---- P: C-layout -> A-layout via LDS transpose bounce ----
#pragma unroll
        for (int r = 0; r < 8; ++r)
            pLDS[(r + half * 8) * 16 + l16] = s[r];
        __syncthreads();
        v2f preg[4];
#pragma unroll
        for (int kb = 0; kb < 4; ++kb) {
            preg[kb].x = pLDS[l16 * 16 + kb * 4 + half * 2];
            preg[kb].y = pLDS[l16 * 16 + kb * 4 + half * 2 + 1];
        }
        __syncthreads();

        // ---- O += P @ V : 4 d-tiles of 16, K-dim = 16 (4 WMMAs each) ----
#pragma unroll
        for (int t = 0; t < 4; ++t) {
#pragma unroll
            for (int kb = 0; kb < 4; ++kb) {
                const float* vp = kv_b + (size_t)(j0 + kb * 4 + half * 2) * 128
                                + 64 + t * 16 + l16;
                v2f bv;
                bv.x = vp[0];
                bv.y = vp[128];
                o[t] = __builtin_amdgcn_wmma_f32_16x16x4_f32(
                    false, preg[kb], false, bv, (short)0, o[t], false, false);
            }
        }
    }

    // ---- epilogue: normalize and store to out_tmp[b][i][h*64+d] ----
    float* ob = out_tmp + ((size_t)(b * NN) + i0) * INNER + h * DH;
#pragma unroll
    for (int r = 0; r < 8; ++r) {
        const float inv_l = __frcp_rn(lrow[r]);
        const int row = r + half * 8;
#pragma unroll
        for (int t = 0; t < 4; ++t) {
            ob[(size_t)row * INNER + t * 16 + l16] = o[t][r] * inv_l;
        }
    }
}

// ---------------------------------------------------------------------------
// Launcher
// ---------------------------------------------------------------------------
extern "C" void kernel_launch(void* const* d_in, const int* in_sizes, int n_in,
                              void* d_out, int out_size, void* d_ws, size_t ws_size,
                              hipStream_t stream) {
    const float* x         = (const float*)d_in[0];
    const float* attn_bias = (const float*)d_in[1];
    const float* ln_w      = (const float*)d_in[2];
    const float* ln_b      = (const float*)d_in[3];
    const float* wq        = (const float*)d_in[4];
    const float* wkv       = (const float*)d_in[5];
    const float* wo        = (const float*)d_in[6];
    const int*   mask      = (const int*)d_in[7];
    float* out = (float*)d_out;

    const int M = BB * NN;  // 4096 rows

    // Workspace layout (floats): xn | q | kv | out_tmp  (~52.4 MB total)
    float* ws   = (float*)d_ws;
    float* xn   = ws;                          // M * 1024
    float* qbuf = xn + (size_t)M * DD;         // M * 1024
    float* kvb  = qbuf + (size_t)M * INNER;    // M * 128
    float* otmp = kvb + (size_t)M * 2 * DH;    // M * 1024

    // 1) LayerNorm
    ln_kernel<<<dim3(M), dim3(256), 0, stream>>>(x, ln_w, ln_b, xn);

    // 2) kv = x @ wkv        [4096 x 128]
    gemm16_kernel<<<dim3(M / 16, 1), dim3(256), 0, stream>>>(
        x, wkv, kvb, M, DD, 2 * DH, 1.0f);

    // 3) q = (xn @ wq) * DH^-0.5   [4096 x 1024], row-major == (B,N,H,DH)
    gemm16_kernel<<<dim3(M / 16, INNER / 128), dim3(256), 0, stream>>>(
        xn, wq, qbuf, M, DD, INNER, 0.125f);

    // 4) fused attention -> out_tmp [B,N,INNER]
    attn_kernel<<<dim3(NN / 16, HH, BB), dim3(32), 0, stream>>>(
        qbuf, kvb, attn_bias, mask, otmp);

    // 5) out = out_tmp @ wo   [4096 x 1024]
    gemm16_kernel<<<dim3(M / 16, DD / 128), dim3(256), 0, stream>>>(
        otmp, wo, out, M, INNER, DD, 1.0f);

    (void)in_sizes; (void)n_in; (void)out_size; (void)ws_size;
}